// CosimLoss_33990371181078
// MI455X (gfx1250) — compile-verified
//
#include <hip/hip_runtime.h>
#include <hip/hip_bf16.h>
#include <cstdint>

#define LANES 32            // wave32 on gfx1250
#define WAVES_PER_BLOCK 8
#define THREADS (WAVES_PER_BLOCK * LANES)
#define CHUNK_FLOATS 128    // 512 B per async round per array (32 lanes x 16 B)
#define NBUF 4              // 4-deep ring: up to 6 outstanding async loads/wave

// ---- CDNA5 async global->LDS copy (one b128 per lane, non-temporal) --------
__device__ __forceinline__ void async_ld16(uint32_t lds_off, const float* gaddr) {
    // vdst = LDS byte offset (VGPR), vaddr = 64-bit global address (VGPR pair)
    // Stream is read-once and larger than the 192MB L2 -> non-temporal hint.
    asm volatile("global_load_async_to_lds_b128 %0, %1, off th:TH_LOAD_NT"
                 :: "v"(lds_off), "v"(gaddr)
                 : "memory");
}
__device__ __forceinline__ void wait_async6() { asm volatile("s_wait_asynccnt 6" ::: "memory"); }
__device__ __forceinline__ void wait_async4() { asm volatile("s_wait_asynccnt 4" ::: "memory"); }
__device__ __forceinline__ void wait_async2() { asm volatile("s_wait_asynccnt 2" ::: "memory"); }
__device__ __forceinline__ void wait_async0() { asm volatile("s_wait_asynccnt 0" ::: "memory"); }

__device__ __forceinline__ void accum_chunk(const float* __restrict__ qb,
                                            const float* __restrict__ ab,
                                            int lane,
                                            float& dot, float& qq, float& aa) {
    const float4 q = *(const float4*)&qb[lane * 4];
    const float4 a = *(const float4*)&ab[lane * 4];
    dot = fmaf(q.x, a.x, dot); dot = fmaf(q.y, a.y, dot);
    dot = fmaf(q.z, a.z, dot); dot = fmaf(q.w, a.w, dot);
    qq  = fmaf(q.x, q.x, qq);  qq  = fmaf(q.y, q.y, qq);
    qq  = fmaf(q.z, q.z, qq);  qq  = fmaf(q.w, q.w, qq);
    aa  = fmaf(a.x, a.x, aa);  aa  = fmaf(a.y, a.y, aa);
    aa  = fmaf(a.z, a.z, aa);  aa  = fmaf(a.w, a.w, aa);
}

// ---- Kernel 1: per-row cosine via async 4-deep streaming pipeline ----------
__global__ __launch_bounds__(THREADS)
void cosrow_kernel(const float* __restrict__ ques,
                   const float* __restrict__ ans,
                   float* __restrict__ cos_out,
                   int D) {
    __shared__ __align__(16) float sq[WAVES_PER_BLOCK][NBUF][CHUNK_FLOATS];
    __shared__ __align__(16) float sa[WAVES_PER_BLOCK][NBUF][CHUNK_FLOATS];

    const int wave = threadIdx.x / LANES;
    const int lane = threadIdx.x % LANES;
    const long row = (long)blockIdx.x * WAVES_PER_BLOCK + wave;

    const float* qp = ques + row * (long)D + lane * 4;
    const float* ap = ans  + row * (long)D + lane * 4;

    // LDS byte offsets for this lane's 16B slot in each ring buffer
    uint32_t lq[NBUF], la[NBUF];
#pragma unroll
    for (int b = 0; b < NBUF; ++b) {
        lq[b] = (uint32_t)(uintptr_t)&sq[wave][b][lane * 4];
        la[b] = (uint32_t)(uintptr_t)&sa[wave][b][lane * 4];
    }

    const int nchunks = D / CHUNK_FLOATS;   // D % CHUNK_FLOATS == 0

    // Prime buffers 0..2 (up to 6 outstanding async loads per wave)
    const int nprime = (nchunks < 3) ? nchunks : 3;
    for (int p = 0; p < nprime; ++p) {
        async_ld16(lq[p], qp + (long)p * CHUNK_FLOATS);
        async_ld16(la[p], ap + (long)p * CHUNK_FLOATS);
    }

    const float* qnext = qp + (long)3 * CHUNK_FLOATS;
    const float* anext = ap + (long)3 * CHUNK_FLOATS;

    float dot = 0.0f, qq = 0.0f, aa = 0.0f;

    // Steady state: branch-free; ring index is a compile-time constant
    // within each unroll-by-4 body.
    int c = 0;
#pragma unroll 4
    for (; c + 3 < nchunks; ++c) {
        const int nb = (c + 3) & (NBUF - 1);
        async_ld16(lq[nb], qnext);
        async_ld16(la[nb], anext);
        qnext += CHUNK_FLOATS;
        anext += CHUNK_FLOATS;
        wait_async6();   // 2 loads/chunk, in-order completion: chunk c landed
        const int b = c & (NBUF - 1);
        accum_chunk(sq[wave][b], sa[wave][b], lane, dot, qq, aa);
    }

    // Epilogue: drain the last (up to) 3 primed chunks.
    for (; c < nchunks; ++c) {
        const int rem = nchunks - 1 - c;
        if (rem == 2)      wait_async4();
        else if (rem == 1) wait_async2();
        else               wait_async0();
        const int b = c & (NBUF - 1);
        accum_chunk(sq[wave][b], sa[wave][b], lane, dot, qq, aa);
    }

    // wave32 tree reduction
    for (int off = 16; off > 0; off >>= 1) {
        dot += __shfl_xor(dot, off);
        qq  += __shfl_xor(qq,  off);
        aa  += __shfl_xor(aa,  off);
    }

    if (lane == 0) {
        const float qn = fmaxf(sqrtf(qq), 1e-8f);
        const float an = fmaxf(sqrtf(aa), 1e-8f);
        cos_out[row] = dot / (qn * an);
    }
}

// ---- Kernel 2: deterministic fixed-order masked-mean + hinge ---------------
__global__ __launch_bounds__(1024)
void finalize_kernel(const float* __restrict__ cosv,
                     const int* __restrict__ label,
                     const float* __restrict__ marginp,
                     float* __restrict__ out,
                     int B) {
    __shared__ float s_ts[1024], s_fs[1024], s_tc[1024], s_fc[1024];
    const int tid = threadIdx.x;

    float ts = 0.0f, fs = 0.0f, tc = 0.0f, fc = 0.0f;
    for (int r = tid; r < B; r += 1024) {
        const float c = cosv[r];
        const int   l = label[r];
        if (l == 1)       { ts += c; tc += 1.0f; }
        else if (l == -1) { fs += c; fc += 1.0f; }
    }
    s_ts[tid] = ts; s_fs[tid] = fs; s_tc[tid] = tc; s_fc[tid] = fc;
    __syncthreads();

    for (int stride = 512; stride > 0; stride >>= 1) {
        if (tid < stride) {
            s_ts[tid] += s_ts[tid + stride];
            s_fs[tid] += s_fs[tid + stride];
            s_tc[tid] += s_tc[tid + stride];
            s_fc[tid] += s_fc[tid + stride];
        }
        __syncthreads();
    }

    if (tid == 0) {
        const float true_ave  = s_ts[0] / s_tc[0];
        const float false_ave = s_fs[0] / s_fc[0];
        out[0] = fmaxf(0.0f, marginp[0] - true_ave + false_ave);
    }
}

extern "C" void kernel_launch(void* const* d_in, const int* in_sizes, int n_in,
                              void* d_out, int out_size, void* d_ws, size_t ws_size,
                              hipStream_t stream) {
    const float* ques   = (const float*)d_in[0];
    const float* ans    = (const float*)d_in[1];
    const int*   label  = (const int*)d_in[2];
    const float* margin = (const float*)d_in[3];

    const int B = in_sizes[2];                 // 8192 (label has B elements)
    const int D = in_sizes[0] / B;             // 4096 (multiple of CHUNK_FLOATS)

    float* cosv = (float*)d_ws;                // B floats of scratch

    const int blocks = B / WAVES_PER_BLOCK;
    cosrow_kernel<<<blocks, THREADS, 0, stream>>>(ques, ans, cosv, D);
    finalize_kernel<<<1, 1024, 0, stream>>>(cosv, label, margin, (float*)d_out, B);
}